// Encoder_47270410060157
// MI455X (gfx1250) — compile-verified
//
#include <hip/hip_runtime.h>
#include <hip/hip_bf16.h>

typedef __attribute__((ext_vector_type(2))) float v2f;
typedef __attribute__((ext_vector_type(8))) float v8f;

#define EMB 128

// ---------------------------------------------------------------------------
// deg[v] = 1 (self loop)
__global__ void k_init_deg(float* __restrict__ deg, int n) {
    int i = blockIdx.x * blockDim.x + threadIdx.x;
    if (i < n) deg[i] = 1.0f;
}

// deg[col[e]] += 1 for every edge target
__global__ void k_count_deg(const int* __restrict__ colI, float* __restrict__ deg, int E) {
    int i = blockIdx.x * blockDim.x + threadIdx.x;
    if (i < E) atomicAdd(&deg[colI[i]], 1.0f);
}

// deg -> deg^{-1/2}  (deg >= 1 always, so no zero guard needed)
__global__ void k_dinv(float* __restrict__ deg, int n) {
    int i = blockIdx.x * blockDim.x + threadIdx.x;
    if (i < n) deg[i] = rsqrtf(deg[i]);
}

// out = x = item_emb[:n]
__global__ void k_init_x(const float* __restrict__ emb, float* __restrict__ x,
                         float* __restrict__ out, int total) {
    int i = blockIdx.x * blockDim.x + threadIdx.x;
    if (i < total) {
        float v = emb[i];
        x[i]   = v;
        out[i] = v;
    }
}

// ---------------------------------------------------------------------------
// H[mt*16 .. mt*16+15][0..127] = X_tile @ W  via V_WMMA_F32_16X16X4_F32.
// One wave per 16-row stripe; 8 v8f accumulators cover all 128 output cols.
__global__ __launch_bounds__(256) void k_gemm_wmma(const float* __restrict__ X,
                                                   const float* __restrict__ W,
                                                   float* __restrict__ H,
                                                   int mTiles) {
    const int lane = threadIdx.x & 31;
    const int wave = threadIdx.x >> 5;
    const int mt   = blockIdx.x * 8 + wave;
    if (mt >= mTiles) return;           // whole wave exits together (EXEC all-1s for WMMA)

    const int m0   = mt * 16;
    const int half = lane >> 4;         // 0: K pair {0,1}, 1: K pair {2,3}
    const int l16  = lane & 15;
    const int kb   = half * 2;

    const float* __restrict__ xrow = X + (size_t)(m0 + l16) * EMB;

    v8f acc[8];
#pragma unroll
    for (int nt = 0; nt < 8; ++nt) acc[nt] = (v8f){0.f,0.f,0.f,0.f,0.f,0.f,0.f,0.f};

    for (int k = 0; k < EMB; k += 4) {
        // A fragment: lane holds X[m0 + l16][k+kb], X[m0 + l16][k+kb+1]
        v2f a = *(const v2f*)(xrow + k + kb);
#pragma unroll
        for (int nt = 0; nt < 8; ++nt) {
            // B fragment: lane holds W[k+kb][nt*16+l16], W[k+kb+1][nt*16+l16]
            v2f b;
            b.x = W[(size_t)(k + kb    ) * EMB + nt * 16 + l16];
            b.y = W[(size_t)(k + kb + 1) * EMB + nt * 16 + l16];
            acc[nt] = __builtin_amdgcn_wmma_f32_16x16x4_f32(
                false, a, false, b, (short)0, acc[nt], false, false);
        }
    }

    // C/D layout: vgpr i, lane -> row m0 + i + half*8, col nt*16 + l16
#pragma unroll
    for (int nt = 0; nt < 8; ++nt) {
        float* __restrict__ hp = H + (size_t)(m0 + half * 8) * EMB + nt * 16 + l16;
#pragma unroll
        for (int i = 0; i < 8; ++i)
            hp[(size_t)i * EMB] = acc[nt][i];
    }
}

// ---------------------------------------------------------------------------
// agg[v][f] = h[v][f] * dinv[v]^2 + b[f]   (self-loop term + bias, also inits agg)
__global__ void k_self_bias(const float* __restrict__ H, const float* __restrict__ dinv,
                            const float* __restrict__ b, float* __restrict__ agg, int total) {
    int i = blockIdx.x * blockDim.x + threadIdx.x;
    if (i < total) {
        int v = i >> 7;      // /128
        int f = i & 127;
        float d = dinv[v];
        agg[i] = H[i] * d * d + b[f];
    }
}

// agg[c] += h[r] * dinv[r]*dinv[c], one wave per edge, float4 chunk per lane
__global__ __launch_bounds__(256) void k_scatter(const int* __restrict__ rowI,
                                                 const int* __restrict__ colI,
                                                 const float* __restrict__ H,
                                                 const float* __restrict__ dinv,
                                                 float* __restrict__ agg, int E) {
    int gtid = blockIdx.x * blockDim.x + threadIdx.x;
    int e    = gtid >> 5;
    int lane = gtid & 31;
    if (e >= E) return;
    int r = rowI[e];
    int c = colI[e];
    float norm = dinv[r] * dinv[c];
    const float4 hv = *(const float4*)(H + (size_t)r * EMB + lane * 4);
    float* __restrict__ dst = agg + (size_t)c * EMB + lane * 4;
    atomicAdd(dst + 0, hv.x * norm);
    atomicAdd(dst + 1, hv.y * norm);
    atomicAdd(dst + 2, hv.z * norm);
    atomicAdd(dst + 3, hv.w * norm);
}

// out += agg
__global__ void k_accum(float* __restrict__ out, const float* __restrict__ agg, int total) {
    int i = blockIdx.x * blockDim.x + threadIdx.x;
    if (i < total) out[i] += agg[i];
}

// ---------------------------------------------------------------------------
extern "C" void kernel_launch(void* const* d_in, const int* in_sizes, int n_in,
                              void* d_out, int out_size, void* d_ws, size_t ws_size,
                              hipStream_t stream) {
    const int*   edge = (const int*)d_in[0];    // [2, E] int32
    const float* emb  = (const float*)d_in[1];  // [total, 128] f32
    const float* Wall = (const float*)d_in[2];  // [3, 128, 128] f32
    const float* ball = (const float*)d_in[3];  // [3, 128] f32
    float*       out  = (float*)d_out;          // [n, 128] f32

    const int E = in_sizes[0] / 2;
    const int n = out_size / EMB;               // 50001
    const int total = n * EMB;

    const int* rowI = edge;
    const int* colI = edge + E;

    const int mTiles  = (n + 15) / 16;          // 3126
    const int rowsPad = mTiles * 16;            // 50016
    const size_t matElems = (size_t)rowsPad * EMB;

    float* ws   = (float*)d_ws;
    float* bufA = ws;                           // three rotating n x 128 matrices
    float* bufB = ws + matElems;
    float* bufC = ws + 2 * matElems;
    float* dinv = ws + 3 * matElems;            // n floats
    float* bufs[3] = {bufA, bufB, bufC};

    const int T = 256;
    // --- normalization ---
    k_init_deg <<<(n + T - 1) / T, T, 0, stream>>>(dinv, n);
    k_count_deg<<<(E + T - 1) / T, T, 0, stream>>>(colI, dinv, E);
    k_dinv     <<<(n + T - 1) / T, T, 0, stream>>>(dinv, n);
    // --- layer 0 input & output init ---
    k_init_x   <<<(total + T - 1) / T, T, 0, stream>>>(emb, bufs[0], out, total);

    int xi = 0, hi = 1, ai = 2;
    for (int l = 0; l < 3; ++l) {
        k_gemm_wmma<<<(mTiles + 7) / 8, 256, 0, stream>>>(
            bufs[xi], Wall + (size_t)l * EMB * EMB, bufs[hi], mTiles);
        k_self_bias<<<(total + T - 1) / T, T, 0, stream>>>(
            bufs[hi], dinv, ball + (size_t)l * EMB, bufs[ai], total);
        k_scatter<<<((size_t)E * 32 + T - 1) / T, T, 0, stream>>>(
            rowI, colI, bufs[hi], dinv, bufs[ai], E);
        k_accum<<<(total + T - 1) / T, T, 0, stream>>>(out, bufs[ai], total);
        // next layer: x <- agg, reuse old x buffer as next agg
        int t = xi; xi = ai; ai = t;
    }
}